// ChebConvGAD_Hetero_36043365548318
// MI455X (gfx1250) — compile-verified
//
#include <hip/hip_runtime.h>
#include <hip/hip_bf16.h>
#include <cstddef>

typedef float v2f __attribute__((ext_vector_type(2)));
typedef float v8f __attribute__((ext_vector_type(8)));

#define BM 256          // 8 waves x 2 strips x 16 rows
#define BN 64
#define KC 16
#define APAD 4
#define BROW (2 * BN + 4)  // floats per K-pair row of Bs

// ---------------------------------------------------------------------------
// Tiled fp32 WMMA GEMM:  C[M,Nn] (ldc) = epilogue(A[M,K] (lda, optional row
// gather) * B[K,Nn] (ldb) + bias)  ; ACC=true does C += (no bias/epilogue)
// EP: 0 = none, 1 = relu, 2 = leaky_relu(0.01)
// B is staged in LDS K-pair interleaved so each 4x16 B fragment is a single
// aligned ds_load_b64; A rows are K-contiguous (ds_load_b64 as well).
// ---------------------------------------------------------------------------
template <int EP, bool ACC>
__global__ __launch_bounds__(256) void gemm_wmma_f32(
    const float* __restrict__ A, const int* __restrict__ rowidx,
    const float* __restrict__ B, const float* __restrict__ bias,
    float* __restrict__ C, int M, int Nn, int K, int lda, int ldb, int ldc) {
  __shared__ float As[BM][KC + APAD];   // 256*20*4  = 20 KB
  __shared__ float Bs[KC / 2][BROW];    // 8*132*4   = ~4.2 KB

  const int tid  = threadIdx.x;
  const int lane = tid & 31;
  const int wave = tid >> 5;
  const int l    = lane & 15;
  const int hi   = lane >> 4;

  const int mblk = blockIdx.y * BM;
  const int nblk = blockIdx.x * BN;
  const int wm   = wave * 16;           // strip 0 rows; strip 1 at +128

  v8f acc[8];
#pragma unroll
  for (int t = 0; t < 8; ++t) acc[t] = (v8f){0, 0, 0, 0, 0, 0, 0, 0};

  for (int k0 = 0; k0 < K; k0 += KC) {
    __syncthreads();
    // Cooperative A chunk load: BM x KC (16 elems/thread)
    for (int e = tid; e < BM * KC; e += 256) {
      int r = e / KC, kk = e % KC;
      int gr = mblk + r;
      float v = 0.0f;
      if (gr < M && (k0 + kk) < K) {
        int rr = rowidx ? rowidx[gr] : gr;
        v = A[(size_t)rr * lda + (size_t)(k0 + kk)];
      }
      As[r][kk] = v;
    }
    // Cooperative B chunk load: KC x BN, stored K-pair interleaved
    for (int e = tid; e < KC * BN; e += 256) {
      int kk = e / BN, c = e % BN;
      int gc = nblk + c;
      float v = 0.0f;
      if (gc < Nn && (k0 + kk) < K) v = B[(size_t)(k0 + kk) * ldb + gc];
      Bs[kk >> 1][2 * c + (kk & 1)] = v;
    }
    __syncthreads();

#pragma unroll
    for (int kk = 0; kk < KC; kk += 4) {
      // A fragments 16x4 (lanes 0-15: K={kk,kk+1}; lanes 16-31: K={kk+2,kk+3})
      v2f a0 = *(const v2f*)&As[wm + l][kk + 2 * hi];
      v2f a1 = *(const v2f*)&As[wm + 128 + l][kk + 2 * hi];
      // B fragments 4x16: pair-row p = kk/2 + hi holds K={kk+2hi, kk+2hi+1}
      const float* brow = &Bs[(kk >> 1) + hi][0];
#pragma unroll
      for (int t = 0; t < 4; ++t) {
        v2f b = *(const v2f*)&brow[2 * (t * 16 + l)];
        acc[t] = __builtin_amdgcn_wmma_f32_16x16x4_f32(
            false, a0, false, b, (short)0, acc[t], false, false);
        acc[4 + t] = __builtin_amdgcn_wmma_f32_16x16x4_f32(
            false, a1, false, b, (short)0, acc[4 + t], false, false);
      }
    }
  }

  // Epilogue. D layout: VGPR vv -> row (vv + 8*hi), col = lane&15
#pragma unroll
  for (int s = 0; s < 2; ++s) {
#pragma unroll
    for (int vv = 0; vv < 8; ++vv) {
      int r = mblk + wm + s * 128 + vv + 8 * hi;
      if (r >= M) continue;
#pragma unroll
      for (int t = 0; t < 4; ++t) {
        int c = nblk + t * 16 + l;
        if (c >= Nn) continue;
        float val = acc[4 * s + t][vv];
        if (bias) val += bias[c];
        if (EP == 1) val = val > 0.0f ? val : 0.0f;
        if (EP == 2) val = val > 0.0f ? val : 0.01f * val;
        size_t off = (size_t)r * ldc + c;
        if (ACC)
          C[off] += val;
        else
          C[off] = val;
      }
    }
  }
}

// ---------------------------------------------------------------------------
__global__ void zero_f32(float* __restrict__ p, size_t n) {
  size_t i = (size_t)blockIdx.x * blockDim.x + threadIdx.x;
  if (i < n) p[i] = 0.0f;
}

// G[dst[e], j] += norm[src[e]] * x[row(src[e]), j]   (one edge per block)
__global__ void scatter_edges(const float* __restrict__ x, int ldx,
                              const int* __restrict__ rowidx,
                              const int* __restrict__ src,
                              const int* __restrict__ dst,
                              const float* __restrict__ norm,
                              float* __restrict__ G, int E, int Hh) {
  int e = blockIdx.x;
  int j = threadIdx.x;
  if (e >= E || j >= Hh) return;
  int s = src[e], d = dst[e];
  float ns = norm[s];
  int rr = rowidx ? rowidx[s] : s;
  float v = ns * x[(size_t)rr * ldx + j];
  atomicAdd(&G[(size_t)d * Hh + j], v);
}

// C[i, 0:H] = x[row(i)];  C[i, H:2H] = (re-1)*x - re*norm[i]*G[i]
__global__ void build_cat(const float* __restrict__ x, int ldx,
                          const int* __restrict__ rowidx,
                          const float* __restrict__ G,
                          const float* __restrict__ norm,
                          const float* __restrict__ lam, float* __restrict__ C,
                          int M, int Hh) {
  size_t idx = (size_t)blockIdx.x * blockDim.x + threadIdx.x;
  size_t total = (size_t)M * Hh;
  if (idx >= total) return;
  int i = (int)(idx / Hh);
  int j = (int)(idx % Hh);
  float re = 2.0f / lam[0];
  int rr = rowidx ? rowidx[i] : i;
  float xv = x[(size_t)rr * ldx + j];
  float agg = G[(size_t)i * Hh + j] * norm[i];
  size_t base = (size_t)i * (2 * Hh);
  C[base + j] = xv;
  C[base + Hh + j] = (re - 1.0f) * xv - re * agg;
}

__global__ void accum_f32(float* __restrict__ dst, const float* __restrict__ s,
                          size_t n) {
  size_t i = (size_t)blockIdx.x * blockDim.x + threadIdx.x;
  if (i < n) dst[i] += s[i];
}

// out[i, c] = b4[c] + sum_j leaky(Bacc[i, j]) * W4[j, c]   (C = 2)
__global__ void final_proj(const float* __restrict__ Bacc,
                           const float* __restrict__ W4,
                           const float* __restrict__ b4,
                           float* __restrict__ out, int N, int Hh) {
  int i = blockIdx.x * blockDim.x + threadIdx.x;
  if (i >= N) return;
  float a0 = b4[0], a1 = b4[1];
  const float* row = Bacc + (size_t)i * Hh;
  for (int j = 0; j < Hh; ++j) {
    float v = row[j];
    v = v > 0.0f ? v : 0.01f * v;
    a0 += v * W4[2 * j + 0];
    a1 += v * W4[2 * j + 1];
  }
  out[2 * i + 0] = a0;
  out[2 * i + 1] = a1;
}

// ---------------------------------------------------------------------------
extern "C" void kernel_launch(void* const* d_in, const int* in_sizes, int n_in,
                              void* d_out, int out_size, void* d_ws,
                              size_t ws_size, hipStream_t stream) {
  const int H = 256;
  const int N = in_sizes[0] / H;

  const float* in_feat = (const float*)d_in[0];
  const float* W_lin = (const float*)d_in[19];
  const float* b_lin = (const float*)d_in[20];
  const float* W2    = (const float*)d_in[21];
  const float* b2    = (const float*)d_in[22];
  const float* W1sg  = (const float*)d_in[23];
  const float* b1sg  = (const float*)d_in[24];
  const float* W2sg  = (const float*)d_in[25];
  const float* b2sg  = (const float*)d_in[26];
  const float* Wc1   = (const float*)d_in[27];
  const float* bc1   = (const float*)d_in[28];
  const float* Wc2   = (const float*)d_in[29];
  const float* bc2   = (const float*)d_in[30];
  const float* W3    = (const float*)d_in[31];
  const float* b3    = (const float*)d_in[32];
  const float* W4    = (const float*)d_in[33];
  const float* b4    = (const float*)d_in[34];

  float* ws = (float*)d_ws;
  const size_t NH = (size_t)N * H;
  float* h    = ws;               // [N,H]
  float* hn   = ws + NH;          // [N,H]
  float* Bacc = ws + 2 * NH;      // [N,H]
  float* G    = ws + 3 * NH;      // [N,H]  agg / h11 scratch
  float* P    = ws + 4 * NH;      // [N,H]  h01
  float* Cc   = ws + 5 * NH;      // [N,2H] concat buffer / iso scratch

  dim3 blk(256);
  auto gemm = [&](int ep, bool accf, const float* A, const int* ridx,
                  const float* B, const float* bias, float* Co, int M, int Nn,
                  int K, int lda, int ldb, int ldc) {
    dim3 grid((Nn + BN - 1) / BN, (M + BM - 1) / BM);
    if (accf) {
      gemm_wmma_f32<0, true><<<grid, blk, 0, stream>>>(A, ridx, B, bias, Co, M,
                                                       Nn, K, lda, ldb, ldc);
    } else if (ep == 1) {
      gemm_wmma_f32<1, false><<<grid, blk, 0, stream>>>(A, ridx, B, bias, Co, M,
                                                        Nn, K, lda, ldb, ldc);
    } else if (ep == 2) {
      gemm_wmma_f32<2, false><<<grid, blk, 0, stream>>>(A, ridx, B, bias, Co, M,
                                                        Nn, K, lda, ldb, ldc);
    } else {
      gemm_wmma_f32<0, false><<<grid, blk, 0, stream>>>(A, ridx, B, bias, Co, M,
                                                        Nn, K, lda, ldb, ldc);
    }
  };
  auto zero = [&](float* p, size_t n) {
    zero_f32<<<(unsigned)((n + 255) / 256), blk, 0, stream>>>(p, n);
  };

  // h = leaky(leaky(in_feat @ W_lin + b_lin) @ W2 + b2)
  gemm(2, false, in_feat, nullptr, W_lin, b_lin, hn, N, H, H, H, H, H);
  gemm(2, false, hn, nullptr, W2, b2, h, N, H, H, H, H, H);

  zero(Bacc, NH);

  for (int r = 0; r < 3; ++r) {
    const int* keep = (const int*)d_in[1 + 6 * r];
    const int* iso  = (const int*)d_in[2 + 6 * r];
    const int* src  = (const int*)d_in[3 + 6 * r];
    const int* dst  = (const int*)d_in[4 + 6 * r];
    const float* norm = (const float*)d_in[5 + 6 * r];
    const float* lam  = (const float*)d_in[6 + 6 * r];
    const int M_ = in_sizes[1 + 6 * r];
    const int I_ = in_sizes[2 + 6 * r];
    const int E_ = in_sizes[3 + 6 * r];
    const size_t MH = (size_t)M_ * H;

    // ---- cheb1: h01 = relu([hk, X1(hk)] @ Wc1 + bc1)
    zero(G, MH);
    scatter_edges<<<E_, blk, 0, stream>>>(h, H, keep, src, dst, norm, G, E_, H);
    build_cat<<<(unsigned)((MH + 255) / 256), blk, 0, stream>>>(
        h, H, keep, G, norm, lam, Cc, M_, H);
    gemm(1, false, Cc, nullptr, Wc1, bc1, P, M_, H, 2 * H, 2 * H, H, H);

    // ---- cheb2 prep on h01
    zero(G, MH);
    scatter_edges<<<E_, blk, 0, stream>>>(P, H, nullptr, src, dst, norm, G, E_, H);
    build_cat<<<(unsigned)((MH + 255) / 256), blk, 0, stream>>>(
        P, H, nullptr, G, norm, lam, Cc, M_, H);

    // ---- keep-part of new h: hn[0:M_] = h01 @ W3_top + b3 + h11 @ W3_bot
    gemm(0, false, P, nullptr, W3, b3, hn, M_, H, H, H, H, H);
    gemm(1, false, Cc, nullptr, Wc2, bc2, G, M_, H, 2 * H, 2 * H, H, H);  // h11
    gemm(0, true, G, nullptr, W3 + (size_t)H * H, nullptr, hn, M_, H, H, H, H, H);

    // ---- iso branch: h02 = h[iso] @ W1sg + b1sg ; h12 = h02 @ W2sg + b2sg
    float* S1 = Cc;
    float* S2 = Cc + NH;
    gemm(0, false, h, iso, W1sg, b1sg, S1, I_, H, H, H, H, H);
    gemm(0, false, S1, nullptr, W2sg, b2sg, S2, I_, H, H, H, H, H);
    gemm(0, false, S1, nullptr, W3, b3, hn + MH, I_, H, H, H, H, H);
    gemm(0, true, S2, nullptr, W3 + (size_t)H * H, nullptr, hn + MH, I_, H, H,
         H, H, H);

    // ---- accumulate h_all sum, then h <- hn
    accum_f32<<<(unsigned)((NH + 255) / 256), blk, 0, stream>>>(Bacc, hn, NH);
    float* t = h;
    h = hn;
    hn = t;
  }

  final_proj<<<(N + 255) / 256, blk, 0, stream>>>(Bacc, W4, b4, (float*)d_out,
                                                  N, H);
  (void)n_in;
  (void)out_size;
  (void)ws_size;
}